// r_CS_general_80384607912522
// MI455X (gfx1250) — compile-verified
//
#include <hip/hip_runtime.h>
#include <hip/hip_bf16.h>

typedef __attribute__((ext_vector_type(2))) float v2f;
typedef __attribute__((ext_vector_type(8))) float v8f;

#define M_DIM   16384
#define N_DIM   8192
#define NNZ_DIM 1048576
#define ETA_INV (1.0f / 1000000.0f)

// ---------------------------------------------------------------------------
// Kernel 1: zero the scratch accumulators and the output scalar.
// ---------------------------------------------------------------------------
__global__ void zero_kernel(float* Ax, float* ATy, float* qx, float* out) {
    int i = blockIdx.x * blockDim.x + threadIdx.x;
    if (i < M_DIM) Ax[i] = 0.0f;
    if (i < N_DIM) { ATy[i] = 0.0f; qx[i] = 0.0f; }
    if (i == 0) out[0] = 0.0f;
}

// ---------------------------------------------------------------------------
// Kernel 2: COO scatter for Ax = A@x and ATy = A^T@y (f32 global atomics).
// ---------------------------------------------------------------------------
__global__ void scatter_kernel(const int* __restrict__ rows,
                               const int* __restrict__ cols,
                               const float* __restrict__ vals,
                               const float* __restrict__ x,
                               const float* __restrict__ y,
                               float* __restrict__ Ax,
                               float* __restrict__ ATy) {
    int i = blockIdx.x * blockDim.x + threadIdx.x;
    if (i >= NNZ_DIM) return;
    int r = rows[i];
    int c = cols[i];
    float v = vals[i];
    atomicAdd(&Ax[r],  v * x[c]);
    atomicAdd(&ATy[c], v * y[r]);
}

// ---------------------------------------------------------------------------
// Kernel 3: qx = Q @ x via V_WMMA_F32_16X16X4_F32.
//   grid  = N/16 row-tiles, block = 256 threads = 8 waves.
//   Each wave owns a K-slice of 1024 (256 WMMAs), B columns broadcast x.
//   A layout (16x4 f32): lanes 0-15 -> rows, K={0,1}; lanes 16-31 -> K={2,3}.
//   D layout: lane 0 holds D[0..7][0], lane 16 holds D[8..15][0].
// ---------------------------------------------------------------------------
__global__ void __launch_bounds__(256)
gemv_wmma_kernel(const float* __restrict__ Q,
                 const float* __restrict__ x,
                 float* __restrict__ qx) {
    const int tile = blockIdx.x;            // 16-row tile of Q
    const int tid  = threadIdx.x;
    const int wave = tid >> 5;
    const int lane = tid & 31;
    const int row  = (tile << 4) + (lane & 15);
    const int koff = (lane >> 4) << 1;      // 0 for lanes 0-15, 2 for 16-31

    const int KCHUNK = N_DIM / 8;           // 1024 columns per wave
    const int k0 = wave * KCHUNK;

    const float* __restrict__ qrow = Q + (size_t)row * N_DIM;

    v8f acc = {};
    for (int k = k0; k < k0 + KCHUNK; k += 32) {
#pragma unroll
        for (int u = 0; u < 8; ++u) {
            const int kk = k + (u << 2) + koff;
            v2f a  = *(const v2f*)(qrow + kk);   // A tile: Q[row][kk], Q[row][kk+1]
            v2f bx = *(const v2f*)(x + kk);      // B broadcast: x[kk], x[kk+1]
            acc = __builtin_amdgcn_wmma_f32_16x16x4_f32(
                /*neg_a=*/false, a, /*neg_b=*/false, bx,
                /*c_mod=*/(short)0, acc, /*reuse_a=*/false, /*reuse_b=*/false);
        }
    }

    // Cross-wave reduction: lanes 0/16 carry rows 0-7 / 8-15 of the tile.
    __shared__ float red[8][16];
    if (lane == 0) {
#pragma unroll
        for (int v = 0; v < 8; ++v) red[wave][v] = acc[v];
    } else if (lane == 16) {
#pragma unroll
        for (int v = 0; v < 8; ++v) red[wave][8 + v] = acc[v];
    }
    __syncthreads();
    if (tid < 16) {
        float s = 0.0f;
#pragma unroll
        for (int w = 0; w < 8; ++w) s += red[w][tid];
        qx[(tile << 4) + tid] = s;
    }
}

// ---------------------------------------------------------------------------
// Kernel 4: residual terms + global reduction into the scalar output.
// ---------------------------------------------------------------------------
__global__ void __launch_bounds__(256)
finalize_kernel(const float* __restrict__ Ax, const float* __restrict__ ATy,
                const float* __restrict__ qx,
                const float* __restrict__ b, const float* __restrict__ c,
                const float* __restrict__ x, const float* __restrict__ y,
                const float* __restrict__ Iy, const float* __restrict__ il,
                const float* __restrict__ iu, const float* __restrict__ l,
                const float* __restrict__ u, float* __restrict__ out) {
    int i = blockIdx.x * blockDim.x + threadIdx.x;
    float s = 0.0f;
    if (i < M_DIM) {
        s += fabsf(y[i] * (Ax[i] - b[i]) * Iy[i]);
    }
    if (i < N_DIM) {
        float pg = c[i] - ATy[i] + qx[i];
        float rp = pg > 0.0f ? pg : 0.0f;     // relu(pg)
        float rn = pg < 0.0f ? -pg : 0.0f;    // relu(-pg)
        s += fabsf((x[i] - l[i]) * rp * il[i]);
        s += fabsf((u[i] - x[i]) * rn * iu[i]);
    }
    __shared__ float sdata[256];
    sdata[threadIdx.x] = s;
    __syncthreads();
    for (int off = 128; off > 0; off >>= 1) {
        if (threadIdx.x < off) sdata[threadIdx.x] += sdata[threadIdx.x + off];
        __syncthreads();
    }
    if (threadIdx.x == 0) atomicAdd(out, sdata[0] * ETA_INV);
}

// ---------------------------------------------------------------------------
extern "C" void kernel_launch(void* const* d_in, const int* in_sizes, int n_in,
                              void* d_out, int out_size, void* d_ws, size_t ws_size,
                              hipStream_t stream) {
    const int*   A_rows = (const int*)d_in[0];
    const int*   A_cols = (const int*)d_in[1];
    const float* A_vals = (const float*)d_in[2];
    const float* Q      = (const float*)d_in[3];
    const float* b      = (const float*)d_in[4];
    const float* c      = (const float*)d_in[5];
    const float* x      = (const float*)d_in[6];
    const float* y      = (const float*)d_in[7];
    const float* Iy     = (const float*)d_in[8];
    const float* il     = (const float*)d_in[9];
    const float* iu     = (const float*)d_in[10];
    const float* l      = (const float*)d_in[11];
    const float* u      = (const float*)d_in[12];
    float* out = (float*)d_out;

    // Workspace: Ax (M) | ATy (N) | qx (N)  -> 128 KB total
    float* Ax  = (float*)d_ws;
    float* ATy = Ax + M_DIM;
    float* qx  = ATy + N_DIM;

    zero_kernel<<<(M_DIM + 255) / 256, 256, 0, stream>>>(Ax, ATy, qx, out);
    scatter_kernel<<<(NNZ_DIM + 255) / 256, 256, 0, stream>>>(
        A_rows, A_cols, A_vals, x, y, Ax, ATy);
    gemv_wmma_kernel<<<N_DIM / 16, 256, 0, stream>>>(Q, x, qx);
    finalize_kernel<<<(M_DIM + 255) / 256, 256, 0, stream>>>(
        Ax, ATy, qx, b, c, x, y, Iy, il, iu, l, u, out);
}